// DualAttentionBlock_53300544144075
// MI455X (gfx1250) — compile-verified
//
#include <hip/hip_runtime.h>
#include <hip/hip_bf16.h>
#include <math.h>

typedef __attribute__((ext_vector_type(2))) float v2f;
typedef __attribute__((ext_vector_type(8))) float v8f;

#define NB 2
#define NC 64
#define NH 48
#define NW 48
#define NL 48
#define SP (NH*NW*NL)                 // 110592 spatial per (b,c)
#define HWP (NH*NW)                   // 2304
#define GN 24
#define SG (GN*GN*GN)                 // 13824
#define TOT ((size_t)NB*NC*SP)        // 14155776 elements of x / theta_x
#define TCD_PER ((size_t)HWP*NC*NC)   // 9437184 per-batch elements of theta_c_d

// ---------------------------------------------------------------------------
// f32 WMMA wrapper: D(16x16) = A(16x4) * B(4x16) + C
// ---------------------------------------------------------------------------
__device__ inline v8f wmma_f32(v2f a, v2f b, v8f c) {
  return __builtin_amdgcn_wmma_f32_16x16x4_f32(
      /*neg_a=*/false, a, /*neg_b=*/false, b,
      /*c_mod=*/(short)0, c, /*reuse_a=*/false, /*reuse_b=*/false);
}

// ---------------------------------------------------------------------------
// K1/K2: 1x1x1 conv == GEMM: out[b,o,s] = sum_c W[o,c] * in[b,c,s] + bias[o]
// S is a compile-time constant so all strides fold into immediate offsets.
// Block = 128 threads = 4 waves; wave = M-tile of 16 output channels.
// Each wave processes 4 consecutive spatial 16-tiles reusing the
// register-resident A (weight) tile.  blockIdx.x = group of 4 N-tiles.
// ---------------------------------------------------------------------------
template<int S>
__global__ void conv1x1_wmma(const float* __restrict__ in, const float* __restrict__ Wm,
                             const float* __restrict__ bias, float* __restrict__ out) {
  const int ntb  = blockIdx.x * 4;
  const int b    = blockIdx.y;
  const int lane = threadIdx.x & 31;
  const int wave = threadIdx.x >> 5;
  const int l16  = lane & 15, hi = lane >> 4;
  const int m_base = wave * 16;
  const float* inb  = in  + (size_t)b * NC * S;
  float*       outb = out + (size_t)b * NC * S;

  // A tile (16x64 weights for this wave's M rows), resident in registers.
  v2f A[16];
#pragma unroll
  for (int kk = 0; kk < 16; ++kk)
    A[kk] = *(const v2f*)(Wm + (m_base + l16) * NC + kk * 4 + 2 * hi);

  float bv8[8];
#pragma unroll
  for (int r = 0; r < 8; ++r) bv8[r] = bias[m_base + r + 8 * hi];

  for (int t = 0; t < 4; ++t) {
    const int n = (ntb + t) * 16 + l16;
    v8f c;
#pragma unroll
    for (int r = 0; r < 8; ++r) c[r] = bv8[r];
#pragma unroll
    for (int kk = 0; kk < 16; ++kk) {     // K = 64 in chunks of 4
      const int k0 = kk * 4 + 2 * hi;
      v2f bv;
      bv.x = inb[(size_t)k0 * S + n];
      bv.y = inb[(size_t)(k0 + 1) * S + n];
      c = wmma_f32(A[kk], bv, c);
    }
#pragma unroll
    for (int r = 0; r < 8; ++r)
      outb[(size_t)(m_base + r + 8 * hi) * S + n] = c[r];
  }
}

// ---------------------------------------------------------------------------
// K3: trilinear 2x upsample, half-pixel centers (align_corners=False),
// edge-clamped (matches jax.image.resize 'trilinear' for exact scale 2).
// For out index o: in = o/2 - 0.25 -> even o: base=o/2-1, f=0.75; odd: base=o/2, f=0.25
// ---------------------------------------------------------------------------
__device__ inline void lincoord(int o, int& i0, int& i1, float& f1) {
  const int k = o >> 1;
  int base;
  if (o & 1) { base = k;     f1 = 0.25f; }
  else       { base = k - 1; f1 = 0.75f; }
  i0 = base < 0 ? 0 : base;
  i1 = (base + 1 > GN - 1) ? GN - 1 : base + 1;
}

__global__ void upsample2x(const float* __restrict__ src, float* __restrict__ dst) {
  size_t idx = (size_t)blockIdx.x * 256 + threadIdx.x;
  if (idx >= TOT) return;
  int l = (int)(idx % NL); size_t t = idx / NL;
  int w = (int)(t % NW); t /= NW;
  int h = (int)(t % NH); t /= NH;
  int bc = (int)t;
  int h0, h1, w0, w1, l0, l1; float fh, fw, fl;
  lincoord(h, h0, h1, fh);
  lincoord(w, w0, w1, fw);
  lincoord(l, l0, l1, fl);
  const float* s = src + (size_t)bc * SG;
  const float v000 = s[h0*576 + w0*24 + l0], v001 = s[h0*576 + w0*24 + l1];
  const float v010 = s[h0*576 + w1*24 + l0], v011 = s[h0*576 + w1*24 + l1];
  const float v100 = s[h1*576 + w0*24 + l0], v101 = s[h1*576 + w0*24 + l1];
  const float v110 = s[h1*576 + w1*24 + l0], v111 = s[h1*576 + w1*24 + l1];
  const float c00 = v000 + fl * (v001 - v000);
  const float c01 = v010 + fl * (v011 - v010);
  const float c10 = v100 + fl * (v101 - v100);
  const float c11 = v110 + fl * (v111 - v110);
  const float c0  = c00 + fw * (c01 - c00);
  const float c1  = c10 + fw * (c11 - c10);
  dst[idx] = c0 + fh * (c1 - c0);
}

// ---------------------------------------------------------------------------
// K4: per-(b,c) sum and min of tc = theta_x * theta_g_up (one block per row)
// ---------------------------------------------------------------------------
__global__ void reduce_tc_rows(const float* __restrict__ tx, const float* __restrict__ tg,
                               float* __restrict__ sums, float* __restrict__ mins) {
  __shared__ float ss[256];
  __shared__ float sm[256];
  const int bc = blockIdx.x;
  const float* a = tx + (size_t)bc * SP;
  const float* b = tg + (size_t)bc * SP;
  float s = 0.f, m = 3.4e38f;
  for (int i = threadIdx.x; i < SP; i += 256) {
    const float v = a[i] * b[i];
    s += v;
    m = fminf(m, v);
  }
  ss[threadIdx.x] = s; sm[threadIdx.x] = m;
  __syncthreads();
  for (int off = 128; off > 0; off >>= 1) {
    if ((int)threadIdx.x < off) {
      ss[threadIdx.x] += ss[threadIdx.x + off];
      sm[threadIdx.x] = fminf(sm[threadIdx.x], sm[threadIdx.x + off]);
    }
    __syncthreads();
  }
  if (threadIdx.x == 0) { sums[bc] = ss[0]; mins[bc] = sm[0]; }
}

__global__ void final_min128(const float* __restrict__ mins, float* __restrict__ minv) {
  __shared__ float sm[128];
  sm[threadIdx.x] = mins[threadIdx.x];
  __syncthreads();
  for (int off = 64; off > 0; off >>= 1) {
    if ((int)threadIdx.x < off) sm[threadIdx.x] = fminf(sm[threadIdx.x], sm[threadIdx.x + off]);
    __syncthreads();
  }
  if (threadIdx.x == 0) *minv = sm[0];
}

// ---------------------------------------------------------------------------
// K5: p = (tx*tg - minv) / (sum_bc - SP*minv) * x
// ---------------------------------------------------------------------------
__global__ void compute_p(const float* __restrict__ x, const float* __restrict__ tx,
                          const float* __restrict__ tg, const float* __restrict__ sums,
                          const float* __restrict__ minv, float* __restrict__ p) {
  size_t idx = (size_t)blockIdx.x * 256 + threadIdx.x;
  if (idx >= TOT) return;
  const int   bc  = (int)(idx / SP);
  const float mv  = *minv;
  const float den = sums[bc] - (float)SP * mv;
  const float tc  = tx[idx] * tg[idx] - mv;
  p[idx] = tc * x[idx] / den;
}

// ---------------------------------------------------------------------------
// K6: theta_c_d[b,h,w,i,j] = sum_l TX[i,l]*TG[j,l] per position.
// M=i=64, N=j=64, K=l=48.  One 128-thread block (4 waves) per (b,h,w);
// wave = M-tile; A (TX tile) register-resident across all 4 N-tiles.
// Both operands contiguous in l -> v2f (b64) loads.
// ---------------------------------------------------------------------------
__global__ void tcd_gemm(const float* __restrict__ tx, const float* __restrict__ tg,
                         float* __restrict__ tcd) {
  const int pos  = blockIdx.x % HWP;
  const int b    = blockIdx.x / HWP;
  const int lane = threadIdx.x & 31;
  const int wave = threadIdx.x >> 5;
  const int l16  = lane & 15, hi = lane >> 4;
  const int m_base = wave * 16;
  const float* txb = tx + (size_t)b * NC * SP + (size_t)pos * NL;
  const float* tgb = tg + (size_t)b * NC * SP + (size_t)pos * NL;
  float* outb = tcd + ((size_t)(b * HWP + pos)) * (NC * NC);

  v2f A[12];
#pragma unroll
  for (int kk = 0; kk < 12; ++kk)
    A[kk] = *(const v2f*)(txb + (size_t)(m_base + l16) * SP + kk * 4 + 2 * hi);

  for (int nt = 0; nt < 4; ++nt) {
    const int n_base = nt * 16;
    v8f c = {0.f, 0.f, 0.f, 0.f, 0.f, 0.f, 0.f, 0.f};
#pragma unroll
    for (int kk = 0; kk < 12; ++kk) {     // K = 48
      const v2f bv = *(const v2f*)(tgb + (size_t)(n_base + l16) * SP + kk * 4 + 2 * hi);
      c = wmma_f32(A[kk], bv, c);
    }
#pragma unroll
    for (int r = 0; r < 8; ++r)
      outb[(m_base + r + 8 * hi) * NC + n_base + l16] = c[r];
  }
}

// ---------------------------------------------------------------------------
// K7/K8: deterministic two-stage reduce of theta_c_d: global min + per-batch sum.
// 2048 blocks x 9216 contiguous elements (batch boundary at block 1024).
// ---------------------------------------------------------------------------
__global__ void reduce_tcd_partial(const float* __restrict__ tcd, float* __restrict__ psum,
                                   float* __restrict__ pmin) {
  __shared__ float ss[256];
  __shared__ float sm[256];
  const size_t start = (size_t)blockIdx.x * 9216;
  float s = 0.f, m = 3.4e38f;
  for (int i = threadIdx.x; i < 9216; i += 256) {
    const float v = tcd[start + i];
    s += v;
    m = fminf(m, v);
  }
  ss[threadIdx.x] = s; sm[threadIdx.x] = m;
  __syncthreads();
  for (int off = 128; off > 0; off >>= 1) {
    if ((int)threadIdx.x < off) {
      ss[threadIdx.x] += ss[threadIdx.x + off];
      sm[threadIdx.x] = fminf(sm[threadIdx.x], sm[threadIdx.x + off]);
    }
    __syncthreads();
  }
  if (threadIdx.x == 0) { psum[blockIdx.x] = ss[0]; pmin[blockIdx.x] = sm[0]; }
}

__global__ void final_tcd(const float* __restrict__ psum, const float* __restrict__ pmin,
                          float* __restrict__ minv2, float* __restrict__ sumb) {
  __shared__ float sh[256];
  const int tid = threadIdx.x;
  float m = 3.4e38f;
  for (int i = tid; i < 2048; i += 256) m = fminf(m, pmin[i]);
  sh[tid] = m;
  __syncthreads();
  for (int off = 128; off > 0; off >>= 1) {
    if (tid < off) sh[tid] = fminf(sh[tid], sh[tid + off]);
    __syncthreads();
  }
  if (tid == 0) *minv2 = sh[0];
  __syncthreads();
  for (int b = 0; b < NB; ++b) {
    float s = 0.f;
    for (int i = tid; i < 1024; i += 256) s += psum[b * 1024 + i];
    sh[tid] = s;
    __syncthreads();
    for (int off = 128; off > 0; off >>= 1) {
      if (tid < off) sh[tid] += sh[tid + off];
      __syncthreads();
    }
    if (tid == 0) sumb[b] = sh[0];
    __syncthreads();
  }
}

// ---------------------------------------------------------------------------
// K9: d[b,j,pos,l] = sum_c x[b,c,pos,l] * (tcd[c,j]-minv2)*invden[b]
// M=j=64, K=c=64, N=l=48.  Normalization folded into the register-resident
// A operand (loaded once per wave, reused across all 3 N-tiles).
// ---------------------------------------------------------------------------
__global__ void d_gemm(const float* __restrict__ x, const float* __restrict__ tcd,
                       const float* __restrict__ minv2, const float* __restrict__ sumb,
                       float* __restrict__ dout) {
  const int pos  = blockIdx.x % HWP;
  const int b    = blockIdx.x / HWP;
  const int lane = threadIdx.x & 31;
  const int wave = threadIdx.x >> 5;
  const int l16  = lane & 15, hi = lane >> 4;
  const int m_base = wave * 16;
  const float mv     = *minv2;
  const float invden = 1.0f / (sumb[b] - (float)TCD_PER * mv);
  const float* tcdb = tcd + ((size_t)(b * HWP + pos)) * (NC * NC);
  const float* xb   = x    + (size_t)b * NC * SP + (size_t)pos * NL;
  float*       ob   = dout + (size_t)b * NC * SP + (size_t)pos * NL;

  v2f A[16];
#pragma unroll
  for (int kk = 0; kk < 16; ++kk) {
    const int k0 = kk * 4 + 2 * hi;
    v2f a;
    a.x = (tcdb[k0 * NC + m_base + l16]       - mv) * invden;
    a.y = (tcdb[(k0 + 1) * NC + m_base + l16] - mv) * invden;
    A[kk] = a;
  }

  for (int nt = 0; nt < 3; ++nt) {        // N = 48 -> 3 tiles
    const int n_base = nt * 16;
    v8f c = {0.f, 0.f, 0.f, 0.f, 0.f, 0.f, 0.f, 0.f};
#pragma unroll
    for (int kk = 0; kk < 16; ++kk) {     // K = 64
      const int k0 = kk * 4 + 2 * hi;
      v2f bv;
      bv.x = xb[(size_t)k0 * SP + n_base + l16];
      bv.y = xb[(size_t)(k0 + 1) * SP + n_base + l16];
      c = wmma_f32(A[kk], bv, c);
    }
#pragma unroll
    for (int r = 0; r < 8; ++r)
      ob[(size_t)(m_base + r + 8 * hi) * SP + n_base + l16] = c[r];
  }
}

// ---------------------------------------------------------------------------
// Host launcher
// ---------------------------------------------------------------------------
extern "C" void kernel_launch(void* const* d_in, const int* in_sizes, int n_in,
                              void* d_out, int out_size, void* d_ws, size_t ws_size,
                              hipStream_t stream) {
  const float* x  = (const float*)d_in[0];
  const float* g  = (const float*)d_in[1];
  const float* W1 = (const float*)d_in[2];
  const float* b1 = (const float*)d_in[3];
  const float* W2 = (const float*)d_in[4];
  const float* b2 = (const float*)d_in[5];
  float* out = (float*)d_out;
  float* ws  = (float*)d_ws;

  // workspace layout (floats); theta_g_half overlaps theta_c_d (disjoint lifetimes)
  float* tx    = ws;                               // TOT
  float* tg    = ws + TOT;                         // TOT
  float* tcd   = ws + 2 * TOT;                     // 2*TCD_PER
  float* tgh   = tcd;                              // NB*NC*SG (used only pre-K6)
  float* sums  = ws + 2 * TOT + 2 * TCD_PER;       // 128
  float* rmin  = sums + NB * NC;                   // 128
  float* minv  = rmin + NB * NC;                   // 1
  float* psum  = minv + 1;                         // 2048
  float* pmin  = psum + 2048;                      // 2048
  float* minv2 = pmin + 2048;                      // 1
  float* sumb  = minv2 + 1;                        // 2

  // 1-2: channel-mix GEMMs (WMMA f32, compile-time strides)
  conv1x1_wmma<SP><<<dim3(SP / 64, NB), 128, 0, stream>>>(x, W1, b1, tx);
  conv1x1_wmma<SG><<<dim3(SG / 64, NB), 128, 0, stream>>>(g, W2, b2, tgh);
  // 3: trilinear upsample gate
  upsample2x<<<(unsigned)(TOT / 256), 256, 0, stream>>>(tgh, tg);
  // 4-5: gate normalization stats
  reduce_tc_rows<<<NB * NC, 256, 0, stream>>>(tx, tg, sums, rmin);
  final_min128<<<1, 128, 0, stream>>>(rmin, minv);
  // 6: p output (first half of d_out)
  compute_p<<<(unsigned)(TOT / 256), 256, 0, stream>>>(x, tx, tg, sums, minv, out);
  // 7: per-position C x C attention maps (WMMA f32)
  tcd_gemm<<<NB * HWP, 128, 0, stream>>>(tx, tg, tcd);
  // 8-9: attention-map normalization stats
  reduce_tcd_partial<<<2048, 256, 0, stream>>>(tcd, psum, pmin);
  final_tcd<<<1, 256, 0, stream>>>(psum, pmin, minv2, sumb);
  // 10: d output (second half of d_out), normalization folded into A operand
  d_gemm<<<NB * HWP, 128, 0, stream>>>(x, tcd, minv2, sumb, out + TOT);
}